// BiMPMatching_54451595379140
// MI455X (gfx1250) — compile-verified
//
#include <hip/hip_runtime.h>
#include <math.h>

#define BB 8
#define SS 256
#define HH 128
#define CSTR 256   // context inner stride (H2)
#define PP 20
#define NPERS 105
#define EPSF 1e-7f
#define NEGF -1000000000.0f

typedef __attribute__((ext_vector_type(2))) float v2f;
typedef __attribute__((ext_vector_type(8))) float v8f;

__device__ __forceinline__ v8f wmma_k4(v2f a, v2f b, v8f c) {
    // D = A(16x4,f32) * B(4x16,f32) + C(16x16,f32)
    return __builtin_amdgcn_wmma_f32_16x16x4_f32(false, a, false, b, (short)0, c, false, false);
}

__device__ __forceinline__ float redsum32(float v) {
#pragma unroll
    for (int m = 16; m >= 1; m >>= 1) v += __shfl_xor(v, m, 32);
    return v;
}
__device__ __forceinline__ float redmax32(float v) {
#pragma unroll
    for (int m = 16; m >= 1; m >>= 1) v = fmaxf(v, __shfl_xor(v, m, 32));
    return v;
}

// ---------------- per-row plain norm + weighted norms (w_maxpool) ----------------
__global__ void k_wnorm(const float* __restrict__ ctx, const float* __restrict__ w,
                        float* __restrict__ nrm, float* __restrict__ wn) {
    int b = blockIdx.y, i = blockIdx.x, lane = threadIdx.x;
    const float* row = ctx + (size_t)(b * SS + i) * CSTR + lane * 4;
    float4 t = *(const float4*)row;
    float s = t.x * t.x + t.y * t.y + t.z * t.z + t.w * t.w;
    s = redsum32(s);
    if (lane == 0) nrm[b * SS + i] = sqrtf(s);
#pragma unroll
    for (int p = 0; p < PP; ++p) {
        float4 wv = *(const float4*)(w + p * HH + lane * 4);
        float ws = t.x * t.x * wv.x * wv.x + t.y * t.y * wv.y * wv.y +
                   t.z * t.z * wv.z * wv.z + t.w * t.w * wv.w * wv.w;
        ws = redsum32(ws);
        if (lane == 0) wn[(b * PP + p) * SS + i] = sqrtf(ws);
    }
}

// ---------------- last valid row per batch + mask popcount ----------------
__global__ void k_lastcnt(const float* __restrict__ ctx, const unsigned char* __restrict__ mask,
                          float* __restrict__ last, float* __restrict__ cnt) {
    int b = blockIdx.x, t = threadIdx.x;
    int c = 0;
    for (int j = 0; j < SS; ++j) c += mask[b * SS + j] ? 1 : 0;
    int idx = c - 1; if (idx < 0) idx = 0;
    last[b * HH + t] = ctx[(size_t)(b * SS + idx) * CSTR + t];
    if (t == 0) cnt[b] = (float)c;
}

// ---------------- cosine matrix via WMMA f32 16x16x4 ----------------
__global__ void k_cos(const float* __restrict__ c1, const float* __restrict__ c2,
                      const float* __restrict__ n1, const float* __restrict__ n2,
                      float* __restrict__ cosm) {
    int b = blockIdx.y;
    int it = blockIdx.x / (SS / 16), jt = blockIdx.x % (SS / 16);
    int lane = threadIdx.x, half = lane >> 4, l = lane & 15;
    const float* arow = c1 + (size_t)(b * SS + it * 16 + l) * CSTR; // M = l
    const float* brow = c2 + (size_t)(b * SS + jt * 16 + l) * CSTR; // N = l
    v8f acc = {};
#pragma unroll
    for (int k = 0; k < HH; k += 4) {
        v2f a  = *(const v2f*)(arow + k + 2 * half);
        v2f bb = *(const v2f*)(brow + k + 2 * half);
        acc = wmma_k4(a, bb, acc);
    }
    float nj = n2[b * SS + jt * 16 + l];
#pragma unroll
    for (int r = 0; r < 8; ++r) {
        int i = it * 16 + r + 8 * half;
        float v = acc[r] / (n1[b * SS + i] * nj + EPSF);
        cosm[(size_t)(b * SS + i) * SS + jt * 16 + l] = v;
    }
}

// ---------------- masked max/mean over one axis of cos -> cols 0,1 ----------------
__global__ void k_cosred(const float* __restrict__ cosm, int rs, int cs,
                         const unsigned char* __restrict__ mask, const float* __restrict__ cnt,
                         float* __restrict__ out) {
    int b = blockIdx.y, i = blockIdx.x, lane = threadIdx.x;
    const float* base = cosm + (size_t)b * SS * SS + (size_t)i * rs;
    float mx = NEGF, sm = 0.f;
    for (int j = lane; j < SS; j += 32) {
        float v = base[(size_t)j * cs];
        bool ok = mask[b * SS + j];
        mx = fmaxf(mx, ok ? v : NEGF);
        sm += ok ? v : 0.f;
    }
    mx = redmax32(mx); sm = redsum32(sm);
    if (lane == 0) {
        float* orow = out + (size_t)(b * SS + i) * NPERS;
        orow[0] = mx;
        orow[1] = sm / (cnt[b] + EPSF);
    }
}

// ---------------- pairwise mp-match (mm) via WMMA, fused masked max/mean over j ----------------
__global__ void k_mm(const float* __restrict__ t1, const float* __restrict__ t2,
                     const float* __restrict__ w,
                     const float* __restrict__ wni_buf, const float* __restrict__ wnj_buf,
                     const unsigned char* __restrict__ maskR, const float* __restrict__ cntR,
                     float* __restrict__ out) {
    int b = blockIdx.y;
    int p  = blockIdx.x / (SS / 16);
    int it = blockIdx.x % (SS / 16);
    int lane = threadIdx.x, half = lane >> 4, l = lane & 15;
    const float* arow = t1 + (size_t)(b * SS + it * 16 + l) * CSTR;
    const float* wrow = w + p * HH;
    float wn_i[8], rmax[8], rsum[8];
#pragma unroll
    for (int r = 0; r < 8; ++r) {
        wn_i[r] = wni_buf[(b * PP + p) * SS + it * 16 + r + 8 * half];
        rmax[r] = NEGF; rsum[r] = 0.f;
    }
    for (int jt = 0; jt < SS / 16; ++jt) {
        const float* brow = t2 + (size_t)(b * SS + jt * 16 + l) * CSTR;
        v8f acc = {};
#pragma unroll
        for (int k = 0; k < HH; k += 4) {
            v2f a  = *(const v2f*)(arow + k + 2 * half);
            v2f wv = *(const v2f*)(wrow + k + 2 * half);
            a.x *= wv.x * wv.x; a.y *= wv.y * wv.y;   // scale A rows by w^2
            v2f bb = *(const v2f*)(brow + k + 2 * half);
            acc = wmma_k4(a, bb, acc);
        }
        int j = jt * 16 + l;
        bool ok = maskR[b * SS + j];
        float wnj = wnj_buf[(b * PP + p) * SS + j];
#pragma unroll
        for (int r = 0; r < 8; ++r) {
            float v = acc[r] / (wn_i[r] * wnj + EPSF);
            rmax[r] = fmaxf(rmax[r], ok ? v : NEGF);
            rsum[r] += ok ? v : 0.f;
        }
    }
#pragma unroll
    for (int r = 0; r < 8; ++r) {
#pragma unroll
        for (int m = 8; m >= 1; m >>= 1) {  // reduce within 16-lane half (over j)
            rmax[r] = fmaxf(rmax[r], __shfl_xor(rmax[r], m, 32));
            rsum[r] += __shfl_xor(rsum[r], m, 32);
        }
    }
    if (l == 0) {
        float inv = 1.f / (cntR[b] + EPSF);
#pragma unroll
        for (int r = 0; r < 8; ++r) {
            int i = it * 16 + r + 8 * half;
            float* orow = out + (size_t)(b * SS + i) * NPERS;
            orow[23 + p] = rmax[r];
            orow[43 + p] = rsum[r] * inv;
        }
    }
}

// ---------------- attention sums: O = cos(^T) @ src  (M x H, K = S) via WMMA ----------------
__global__ void k_attgemm(const float* __restrict__ cosm, int ars, int acs,
                          const float* __restrict__ src, float* __restrict__ outb) {
    int b = blockIdx.y;
    int it = blockIdx.x / (HH / 16), ht = blockIdx.x % (HH / 16);
    int lane = threadIdx.x, half = lane >> 4, l = lane & 15;
    const float* abase = cosm + (size_t)b * SS * SS;
    int m = it * 16 + l;
    v8f acc = {};
    for (int k = 0; k < SS; k += 4) {
        int kk = k + 2 * half;
        v2f a, bb;
        a.x  = abase[(size_t)m * ars + (size_t)kk * acs];
        a.y  = abase[(size_t)m * ars + (size_t)(kk + 1) * acs];
        bb.x = src[(size_t)(b * SS + kk) * CSTR + ht * 16 + l];
        bb.y = src[(size_t)(b * SS + kk + 1) * CSTR + ht * 16 + l];
        acc = wmma_k4(a, bb, acc);
    }
#pragma unroll
    for (int r = 0; r < 8; ++r) {
        int mm_ = it * 16 + r + 8 * half;
        outb[(size_t)(b * SS + mm_) * HH + ht * 16 + l] = acc[r];
    }
}

// ---------------- in-place softmax over H with row mask ----------------
__global__ void k_softmax(float* __restrict__ buf, const unsigned char* __restrict__ maskRow) {
    __shared__ float sh[HH];
    int b = blockIdx.y, i = blockIdx.x, t = threadIdx.x;
    float x = buf[(size_t)(b * SS + i) * HH + t];
    if (!maskRow[b * SS + i]) x = NEGF;
    sh[t] = x; __syncthreads();
    for (int s = HH / 2; s > 0; s >>= 1) { if (t < s) sh[t] = fmaxf(sh[t], sh[t + s]); __syncthreads(); }
    float mx = sh[0]; __syncthreads();
    float e = expf(x - mx);
    sh[t] = e; __syncthreads();
    for (int s = HH / 2; s > 0; s >>= 1) { if (t < s) sh[t] += sh[t + s]; __syncthreads(); }
    buf[(size_t)(b * SS + i) * HH + t] = e / sh[0];
}

// ---------------- att_max: max_j mask[j] ? cos(i,j)*src[j,h] : NEG ----------------
__global__ void k_attmax(const float* __restrict__ cosm, int ars, int acs,
                         const float* __restrict__ src, const unsigned char* __restrict__ maskRed,
                         float* __restrict__ outb) {
    int b = blockIdx.y, i = blockIdx.x, t = threadIdx.x;
    const float* base = cosm + (size_t)b * SS * SS + (size_t)i * ars;
    float mx = NEGF;
    for (int j = 0; j < SS; ++j) {
        float v = base[(size_t)j * acs] * src[(size_t)(b * SS + j) * CSTR + t];
        mx = fmaxf(mx, maskRed[b * SS + j] ? v : NEGF);
    }
    outb[(size_t)(b * SS + i) * HH + t] = mx;
}

// ---------------- mp_match: single cos-sim + P weighted cos-sims per row ----------------
__global__ void k_mp(const float* __restrict__ t1, const float* __restrict__ t2, int t2row,
                     const float* __restrict__ w, float* __restrict__ out, int colbase) {
    int b = blockIdx.y, i = blockIdx.x, lane = threadIdx.x;
    float4 a = *(const float4*)(t1 + (size_t)(b * SS + i) * CSTR + lane * 4);
    const float* t2p = t2row ? (t2 + (size_t)b * HH) : (t2 + (size_t)(b * SS + i) * HH);
    float4 bv = *(const float4*)(t2p + lane * 4);
    float d  = a.x * bv.x + a.y * bv.y + a.z * bv.z + a.w * bv.w;
    float na = a.x * a.x + a.y * a.y + a.z * a.z + a.w * a.w;
    float nb = bv.x * bv.x + bv.y * bv.y + bv.z * bv.z + bv.w * bv.w;
    d = redsum32(d); na = redsum32(na); nb = redsum32(nb);
    float* orow = out + (size_t)(b * SS + i) * NPERS;
    if (lane == 0) orow[colbase] = d / (sqrtf(na) * sqrtf(nb) + EPSF);
#pragma unroll
    for (int p = 0; p < PP; ++p) {
        float4 wv = *(const float4*)(w + p * HH + lane * 4);
        float wx = wv.x * wv.x, wy = wv.y * wv.y, wz = wv.z * wv.z, ww = wv.w * wv.w;
        float s0 = a.x * bv.x * wx + a.y * bv.y * wy + a.z * bv.z * wz + a.w * bv.w * ww;
        float s1 = a.x * a.x * wx + a.y * a.y * wy + a.z * a.z * wz + a.w * a.w * ww;
        float s2 = bv.x * bv.x * wx + bv.y * bv.y * wy + bv.z * bv.z * wz + bv.w * bv.w * ww;
        s0 = redsum32(s0); s1 = redsum32(s1); s2 = redsum32(s2);
        if (lane == 0) orow[colbase + 1 + p] = s0 / (sqrtf(s1) * sqrtf(s2) + EPSF);
    }
}

extern "C" void kernel_launch(void* const* d_in, const int* in_sizes, int n_in,
                              void* d_out, int out_size, void* d_ws, size_t ws_size,
                              hipStream_t stream) {
    const float* ctx1 = (const float*)d_in[0];
    const float* ctx2 = (const float*)d_in[1];
    const unsigned char* m1 = (const unsigned char*)d_in[2];
    const unsigned char* m2 = (const unsigned char*)d_in[3];
    const float* wfull = (const float*)d_in[4];
    const float* wmax  = (const float*)d_in[5];
    const float* watt  = (const float*)d_in[6];
    const float* wmatt = (const float*)d_in[7];
    float* out = (float*)d_out;
    float* outv1 = out;
    float* outv2 = out + (size_t)BB * SS * NPERS;

    float* ws = (float*)d_ws;
    float* cosm   = ws; ws += (size_t)BB * SS * SS;
    float* n1     = ws; ws += BB * SS;
    float* n2     = ws; ws += BB * SS;
    float* wn1    = ws; ws += BB * PP * SS;
    float* wn2    = ws; ws += BB * PP * SS;
    float* c1last = ws; ws += BB * HH;
    float* c2last = ws; ws += BB * HH;
    float* cnt1   = ws; ws += BB;
    float* cnt2   = ws; ws += BB;
    float* attsum2 = ws; ws += (size_t)BB * SS * HH;  // becomes att_mean_2 after softmax
    float* attsum1 = ws; ws += (size_t)BB * SS * HH;  // becomes att_mean_1 after softmax
    float* attmax2 = ws; ws += (size_t)BB * SS * HH;
    float* attmax1 = ws; ws += (size_t)BB * SS * HH;

    k_wnorm<<<dim3(SS, BB), 32, 0, stream>>>(ctx1, wmax, n1, wn1);
    k_wnorm<<<dim3(SS, BB), 32, 0, stream>>>(ctx2, wmax, n2, wn2);
    k_lastcnt<<<BB, HH, 0, stream>>>(ctx1, m1, c1last, cnt1);
    k_lastcnt<<<BB, HH, 0, stream>>>(ctx2, m2, c2last, cnt2);

    k_cos<<<dim3((SS / 16) * (SS / 16), BB), 32, 0, stream>>>(ctx1, ctx2, n1, n2, cosm);

    k_cosred<<<dim3(SS, BB), 32, 0, stream>>>(cosm, SS, 1, m2, cnt2, outv1);
    k_cosred<<<dim3(SS, BB), 32, 0, stream>>>(cosm, 1, SS, m1, cnt1, outv2);

    k_mm<<<dim3(PP * (SS / 16), BB), 32, 0, stream>>>(ctx1, ctx2, wmax, wn1, wn2, m2, cnt2, outv1);
    k_mm<<<dim3(PP * (SS / 16), BB), 32, 0, stream>>>(ctx2, ctx1, wmax, wn2, wn1, m1, cnt1, outv2);

    k_attgemm<<<dim3((SS / 16) * (HH / 16), BB), 32, 0, stream>>>(cosm, SS, 1, ctx2, attsum2);
    k_attgemm<<<dim3((SS / 16) * (HH / 16), BB), 32, 0, stream>>>(cosm, 1, SS, ctx1, attsum1);
    k_softmax<<<dim3(SS, BB), HH, 0, stream>>>(attsum2, m1);
    k_softmax<<<dim3(SS, BB), HH, 0, stream>>>(attsum1, m2);
    k_attmax<<<dim3(SS, BB), HH, 0, stream>>>(cosm, SS, 1, ctx2, m2, attmax2);
    k_attmax<<<dim3(SS, BB), HH, 0, stream>>>(cosm, 1, SS, ctx1, m1, attmax1);

    k_mp<<<dim3(SS, BB), 32, 0, stream>>>(ctx1, c2last, 1, wfull, outv1, 2);
    k_mp<<<dim3(SS, BB), 32, 0, stream>>>(ctx1, attsum2, 0, watt, outv1, 63);
    k_mp<<<dim3(SS, BB), 32, 0, stream>>>(ctx1, attmax2, 0, wmatt, outv1, 84);
    k_mp<<<dim3(SS, BB), 32, 0, stream>>>(ctx2, c1last, 1, wfull, outv2, 2);
    k_mp<<<dim3(SS, BB), 32, 0, stream>>>(ctx2, attsum1, 0, watt, outv2, 63);
    k_mp<<<dim3(SS, BB), 32, 0, stream>>>(ctx2, attmax1, 0, wmatt, outv2, 84);
}